// SpatiotemporalUncertaintyLoss_32341103739143
// MI455X (gfx1250) — compile-verified
//
#include <hip/hip_runtime.h>
#include <math.h>

typedef float v2f __attribute__((ext_vector_type(2)));
typedef float v4f __attribute__((ext_vector_type(4)));
typedef float v8f __attribute__((ext_vector_type(8)));

#define BIGF      1.0e12f
#define GHOSTF    2.0f
#define W_LIDARF  1.0f
#define W_RADARF  1.0f
#define W_R_TEMPF 1.0f
#define W_R_SPATF 0.1f
// precomputed: 2*log(0.03/10+1e-9), 2*log(0.2/10+1e-9), 2*log(0.1/5+1e-9), 2*log(3/5+1e-9)
#define L_MINF (-11.6182853f)
#define L_MAXF (-7.8240459f)
#define R_MINF (-7.8240459f)
#define R_MAXF (-1.0216512f)

__device__ __forceinline__ float safe_dt_of(const float* dt) {
    float d = dt[0];
    return (d > 0.01f) ? d : 0.1f;
}

// raw v_min_num without NaN-canonicalization max-pairs
__device__ __forceinline__ float fmin_raw(float a, float b) {
    float r;
    asm("v_min_num_f32 %0, %1, %2" : "=v"(r) : "v"(a), "v"(b));
    return r;
}

__device__ __forceinline__ float wave_sum32(float v) {
    #pragma unroll
    for (int m = 1; m <= 16; m <<= 1) v += __shfl_xor(v, m, 32);
    return v;
}

// first index i in [0,n) with a[i] >= key (a sorted ascending); n if none
__device__ __forceinline__ int lower_bound_i(const int* __restrict__ a, int n, int key) {
    int lo = 0, hi = n;
    while (lo < hi) {
        int mid = (lo + hi) >> 1;
        if (a[mid] < key) lo = mid + 1; else hi = mid;
    }
    return lo;
}

// ---------------------------------------------------------------------------
__global__ void zero_kernel(float* __restrict__ p, int n) {
    int i = blockIdx.x * blockDim.x + threadIdx.x;
    if (i < n) p[i] = 0.0f;
}

// Query pack: out4[i] = (a+b, |a+b|^2)
__global__ void pack_sum_kernel(const float* __restrict__ a,
                                const float* __restrict__ b,
                                int n, float* __restrict__ out4) {
    int i = blockIdx.x * blockDim.x + threadIdx.x;
    if (i >= n) return;
    float x = a[i * 3 + 0] + b[i * 3 + 0];
    float y = a[i * 3 + 1] + b[i * 3 + 1];
    float z = a[i * 3 + 2] + b[i * 3 + 2];
    v4f o; o.x = x; o.y = y; o.z = z; o.w = x * x + y * y + z * z;
    *(v4f*)(out4 + i * 4) = o;
}

// Gt pack directly in WMMA-B register order: for column n (tile t=n/16, c=n%16)
//   lane c      (half 0, K=0,1) gets (gx, gy)
//   lane 16+c   (half 1, K=2,3) gets (gz, |g|^2)
// flat: Bpack[(n&~15)*4 + lane*2 + {0,1}]
__global__ void pack_b_kernel(const float* __restrict__ p,
                              int n, float* __restrict__ bpack) {
    int i = blockIdx.x * blockDim.x + threadIdx.x;
    if (i >= n) return;
    float x = p[i * 3 + 0], y = p[i * 3 + 1], z = p[i * 3 + 2];
    float g2 = x * x + y * y + z * z;
    float* base = bpack + (i & ~15) * 4;
    int c = i & 15;
    v2f lohalf; lohalf.x = x; lohalf.y = y;
    v2f hihalf; hihalf.x = z; hihalf.y = g2;
    *(v2f*)(base + c * 2)        = lohalf;
    *(v2f*)(base + (16 + c) * 2) = hihalf;
}

// ---------------------------------------------------------------------------
// WMMA nearest-neighbor with sorted-batch range splitting.
// D = |g|^2 - 2 p.g via V_WMMA_F32_16X16X4_F32; |p|^2 added after the min,
// clamp at 0 after the min (both commute with min). Masked tiles get BIG.
__device__ __forceinline__ void sweep_masked(
        const float* __restrict__ Bpack, const int* __restrict__ Gb,
        int from, int to, int lane, int l16, v2f A,
        const int rb[8], float rmin[8]) {
    for (int colBase = from; colBase < to; colBase += 16) {
        const v2f B = *(const v2f*)(Bpack + colBase * 4 + lane * 2);
        const int cb = Gb[colBase + l16];
        v8f C = {};
        C = __builtin_amdgcn_wmma_f32_16x16x4_f32(
                false, A, false, B, (short)0, C, false, false);
        #pragma unroll
        for (int v = 0; v < 8; ++v) {
            float d = (rb[v] != cb) ? BIGF : C[v];
            rmin[v] = fmin_raw(rmin[v], d);
        }
    }
}

__global__ __launch_bounds__(256) void nn_min_wmma_kernel(
        const float* __restrict__ Q4, const int* __restrict__ Qb, int NQ,
        const float* __restrict__ Bpack, const int* __restrict__ Gb, int NG,
        float* __restrict__ out_min) {
    const int lane = threadIdx.x & 31;
    const int wave = threadIdx.x >> 5;
    const int rowBase = (blockIdx.x * 8 + wave) * 16;
    if (rowBase >= NQ) return;  // wave-uniform: EXEC stays all-ones for WMMA

    const int half = lane >> 4;   // 0: lanes 0-15, 1: lanes 16-31
    const int l16  = lane & 15;

    // A operand, held for the whole sweep
    const v4f q4 = *(const v4f*)(Q4 + (rowBase + l16) * 4);
    v2f A;
    A.x = half ? (-2.0f * q4.z) : (-2.0f * q4.x);   // K = 2*half
    A.y = half ? 1.0f           : (-2.0f * q4.y);   // K = 2*half+1 (pad -> 1)

    // Row batch ids for the 8 C/D rows this lane owns (M = v + 8*half)
    int rb[8];
    #pragma unroll
    for (int v = 0; v < 8; ++v) rb[v] = Qb[rowBase + v + 8 * half];

    // Sorted batches: [lo,hi) = tiles that can match; [midS,midE) = tiles
    // guaranteed single-batch == row batch (mask-free hot loop).
    const int rb0  = Qb[rowBase];
    const int rb15 = Qb[rowBase + 15];
    const int first = lower_bound_i(Gb, NG, rb0);
    const int last  = lower_bound_i(Gb, NG, rb15 + 1);
    int lo = first & ~15;
    int hi = (last + 15) & ~15;  if (hi > NG) hi = NG;
    int midS, midE;
    if (rb0 == rb15) { midS = (first + 15) & ~15; midE = last & ~15; }
    else             { midS = lo;                 midE = lo; }
    if (midS > hi)   midS = hi;
    if (midE < midS) midE = midS;
    // make bounds explicitly scalar (wave-uniform by construction)
    lo   = __builtin_amdgcn_readfirstlane(lo);
    hi   = __builtin_amdgcn_readfirstlane(hi);
    midS = __builtin_amdgcn_readfirstlane(midS);
    midE = __builtin_amdgcn_readfirstlane(midE);

    float rmin[8];
    #pragma unroll
    for (int v = 0; v < 8; ++v) rmin[v] = BIGF;

    // <=1 masked prologue tile (empty for multi-batch rows)
    sweep_masked(Bpack, Gb, lo, midS, lane, l16, A, rb, rmin);

    // hot loop: b64 load + WMMA + 8 raw mins, nothing else
    for (int colBase = midS; colBase < midE; colBase += 16) {
        const v2f B = *(const v2f*)(Bpack + colBase * 4 + lane * 2);
        v8f C = {};
        C = __builtin_amdgcn_wmma_f32_16x16x4_f32(
                false, A, false, B, (short)0, C, false, false);
        #pragma unroll
        for (int v = 0; v < 8; ++v) rmin[v] = fmin_raw(rmin[v], C[v]);
    }

    // <=1 masked epilogue tile (or the full masked sweep remainder)
    sweep_masked(Bpack, Gb, midE, hi, lane, l16, A, rb, rmin);

    // reduce min across the 16 lanes of each half (xor 1,2,4,8 stays in-half)
    #pragma unroll
    for (int v = 0; v < 8; ++v) {
        float x = rmin[v];
        #pragma unroll
        for (int msk = 1; msk <= 8; msk <<= 1)
            x = fmin_raw(x, __shfl_xor(x, msk, 32));
        rmin[v] = x;
    }

    if (l16 == 0) {  // lane 0 writes rows rowBase+0..7, lane 16 writes +8..15
        #pragma unroll
        for (int v = 0; v < 8; ++v) {
            const int row = rowBase + v + 8 * half;
            out_min[row] = fmaxf(rmin[v] + Q4[row * 4 + 3], 0.0f);
        }
    }
}

// ---------------------------------------------------------------------------
// LiDAR finalize: mean(nll + dist_sq) * W_LIDAR, wave-reduced atomic add.
__global__ void lidar_finalize_kernel(const float* __restrict__ log_var,
                                      const float* __restrict__ dmin,
                                      int N, float inv_n,
                                      float* __restrict__ out) {
    int i = blockIdx.x * blockDim.x + threadIdx.x;
    float contrib = 0.0f;
    if (i < N) {
        float lv  = fminf(fmaxf(log_var[i], L_MINF), L_MAXF);
        float dsq = dmin[i];
        float nll = 0.5f * expf(-lv) * dsq + 0.5f * lv;
        contrib = (nll + dsq) * (W_LIDARF * inv_n);
    }
    float s = wave_sum32(contrib);
    if ((threadIdx.x & 31) == 0) atomicAdd(out, s);
}

// ---------------------------------------------------------------------------
// Radar temporal edge scatter: unit motion vectors accumulated at dst.
__global__ void radar_temporal_kernel(const int* __restrict__ ei, int E,
                                      const float* __restrict__ pos,
                                      float* __restrict__ dir,
                                      float* __restrict__ cnt) {
    int e = blockIdx.x * blockDim.x + threadIdx.x;
    if (e >= E) return;
    int src = ei[e];
    int dst = ei[E + e];
    float mx = pos[dst * 3 + 0] - pos[src * 3 + 0];
    float my = pos[dst * 3 + 1] - pos[src * 3 + 1];
    float mz = pos[dst * 3 + 2] - pos[src * 3 + 2];
    float inv = 1.0f / (sqrtf(mx * mx + my * my + mz * mz) + 1e-9f);
    atomicAdd(&dir[dst * 3 + 0], mx * inv);
    atomicAdd(&dir[dst * 3 + 1], my * inv);
    atomicAdd(&dir[dst * 3 + 2], mz * inv);
    atomicAdd(&cnt[dst], 1.0f);
}

// Per-node: normalize mean direction, physics rollout; packed (x,y,z,|p|^2).
__global__ void radar_posphy_kernel(const float* __restrict__ pos,
                                    const float* __restrict__ x4,
                                    const float* __restrict__ dir,
                                    const float* __restrict__ cnt,
                                    const float* __restrict__ dt,
                                    int N, float* __restrict__ posphy4) {
    int i = blockIdx.x * blockDim.x + threadIdx.x;
    if (i >= N) return;
    float c  = fmaxf(cnt[i], 1.0f);
    float dx = dir[i * 3 + 0] / c;
    float dy = dir[i * 3 + 1] / c;
    float dz = dir[i * 3 + 2] / c;
    float inv = 1.0f / (sqrtf(dx * dx + dy * dy + dz * dz) + 1e-9f);
    dx *= inv; dy *= inv; dz *= inv;
    float speed = fabsf(x4[i * 4 + 2]);
    float sdt = safe_dt_of(dt);
    float px = pos[i * 3 + 0] + speed * dx * sdt;
    float py = pos[i * 3 + 1] + speed * dy * sdt;
    float pz = pos[i * 3 + 2] + speed * dz * sdt;
    v4f o; o.x = px; o.y = py; o.z = pz; o.w = px * px + py * py + pz * pz;
    *(v4f*)(posphy4 + i * 4) = o;
}

// Radar->lidar spatial consistency scatter.
__global__ void radar_spatial_kernel(const int* __restrict__ src,
                                     const int* __restrict__ dst, int E,
                                     const float* __restrict__ rpos,
                                     const float* __restrict__ lpos,
                                     float* __restrict__ sum_d,
                                     float* __restrict__ cnt_l) {
    int e = blockIdx.x * blockDim.x + threadIdx.x;
    if (e >= E) return;
    int s = src[e];
    int d = dst[e];
    float dx = rpos[s * 3 + 0] - lpos[d * 3 + 0];
    float dy = rpos[s * 3 + 1] - lpos[d * 3 + 1];
    float dz = rpos[s * 3 + 2] - lpos[d * 3 + 2];
    atomicAdd(&sum_d[s], dx * dx + dy * dy + dz * dz);
    atomicAdd(&cnt_l[s], 1.0f);
}

// Radar finalize: mean(phys/denom)*Wt + mean(spat/denom)*Ws + mean(0.5*lv)
__global__ void radar_finalize_kernel(const float* __restrict__ log_var,
                                      const float* __restrict__ phys_err,
                                      const float* __restrict__ sum_d,
                                      const float* __restrict__ cnt_l,
                                      const float* __restrict__ dt,
                                      int N, float inv_n,
                                      float* __restrict__ out) {
    int i = blockIdx.x * blockDim.x + threadIdx.x;
    float contrib = 0.0f;
    if (i < N) {
        float lv = fminf(fmaxf(log_var[i], R_MINF), R_MAXF);
        float sdt = safe_dt_of(dt);
        float denom = 2.0f * expf(lv) * sdt * sdt + 1e-9f;
        float c = cnt_l[i];
        float cm = fmaxf(c, 1.0f);
        float spat = (c > 0.0f) ? sum_d[i] / (cm * cm) : GHOSTF;
        contrib = (phys_err[i] / denom) * W_R_TEMPF
                + (spat / denom) * W_R_SPATF
                + 0.5f * lv;
        contrib *= W_RADARF * inv_n;
    }
    float s = wave_sum32(contrib);
    if ((threadIdx.x & 31) == 0) atomicAdd(out, s);
}

// ---------------------------------------------------------------------------
extern "C" void kernel_launch(void* const* d_in, const int* in_sizes, int n_in,
                              void* d_out, int out_size, void* d_ws, size_t ws_size,
                              hipStream_t stream) {
    const float* lidar_pos      = (const float*)d_in[0];
    const float* pred_disp      = (const float*)d_in[1];
    const float* log_var_pos    = (const float*)d_in[2];
    const int*   lidar_batch    = (const int*)  d_in[3];
    const float* gt_lidar_pos   = (const float*)d_in[4];
    const int*   gt_lidar_batch = (const int*)  d_in[5];
    const float* gt_radar_pos   = (const float*)d_in[6];
    const int*   gt_radar_batch = (const int*)  d_in[7];
    const float* dt             = (const float*)d_in[8];

    const int N_L  = in_sizes[3];          // 12288
    const int N_GL = in_sizes[5];          // 12288
    const int N_GR = in_sizes[7];          // 8192
    const int N_R  = in_sizes[12];         // 8192
    const int E_T  = in_sizes[13] / 2;     // 65536
    const int E_RL = in_sizes[14];         // 131072

    float* out = (float*)d_out;
    float* ws  = (float*)d_ws;

    // workspace layout (floats); all vec regions 16B aligned
    float* qpack    = ws;                        // 4*N_L
    float* lmin     = qpack + 4 * N_L;           // N_L
    float* gl_bpack = lmin + N_L;                // 4*N_GL
    float* gr_bpack = gl_bpack + 4 * N_GL;       // 4*N_GR
    float* dir      = gr_bpack + 4 * N_GR;       // 3*N_R
    float* cnt_t    = dir + 3 * N_R;             // N_R  (contiguous with dir)
    float* posphy4  = cnt_t + N_R;               // 4*N_R
    float* phys_err = posphy4 + 4 * N_R;         // N_R
    float* sum_d    = phys_err + N_R;            // N_R
    float* cnt_l    = sum_d + N_R;               // N_R  (contiguous with sum_d)

    const int TB = 256;
    auto blocks = [](int n, int tb) { return (n + tb - 1) / tb; };

    // total = 0
    zero_kernel<<<1, 32, 0, stream>>>(out, 1);

    // ---- packs ----
    pack_sum_kernel<<<blocks(N_L, TB), TB, 0, stream>>>(
        lidar_pos, pred_disp, N_L, qpack);
    pack_b_kernel<<<blocks(N_GL, TB), TB, 0, stream>>>(
        gt_lidar_pos, N_GL, gl_bpack);
    pack_b_kernel<<<blocks(N_GR, TB), TB, 0, stream>>>(
        gt_radar_pos, N_GR, gr_bpack);

    // ---- LiDAR term ----
    nn_min_wmma_kernel<<<(N_L / 16 + 7) / 8, 256, 0, stream>>>(
        qpack, lidar_batch, N_L, gl_bpack, gt_lidar_batch, N_GL, lmin);
    lidar_finalize_kernel<<<blocks(N_L, TB), TB, 0, stream>>>(
        log_var_pos, lmin, N_L, 1.0f / (float)N_L, out);

    // ---- Radar terms (r1, r2) reuse the same scratch sequentially ----
    for (int r = 0; r < 2; ++r) {
        const int base = 9 + r * 7;
        const float* rpos  = (const float*)d_in[base + 0];
        const float* rx    = (const float*)d_in[base + 1];
        const float* rlv   = (const float*)d_in[base + 2];
        const int*   rb    = (const int*)  d_in[base + 3];
        const int*   tei   = (const int*)  d_in[base + 4];
        const int*   rlsrc = (const int*)  d_in[base + 5];
        const int*   rldst = (const int*)  d_in[base + 6];

        zero_kernel<<<blocks(4 * N_R, TB), TB, 0, stream>>>(dir, 4 * N_R);
        zero_kernel<<<blocks(2 * N_R, TB), TB, 0, stream>>>(sum_d, 2 * N_R);

        radar_temporal_kernel<<<blocks(E_T, TB), TB, 0, stream>>>(
            tei, E_T, rpos, dir, cnt_t);
        radar_posphy_kernel<<<blocks(N_R, TB), TB, 0, stream>>>(
            rpos, rx, dir, cnt_t, dt, N_R, posphy4);
        nn_min_wmma_kernel<<<(N_R / 16 + 7) / 8, 256, 0, stream>>>(
            posphy4, rb, N_R, gr_bpack, gt_radar_batch, N_GR, phys_err);
        radar_spatial_kernel<<<blocks(E_RL, TB), TB, 0, stream>>>(
            rlsrc, rldst, E_RL, rpos, lidar_pos, sum_d, cnt_l);
        radar_finalize_kernel<<<blocks(N_R, TB), TB, 0, stream>>>(
            rlv, phys_err, sum_d, cnt_l, dt, N_R, 1.0f / (float)N_R, out);
    }
}